// BridgeImportanceHGNN_Simple_3770981286512
// MI455X (gfx1250) — compile-verified
//
#include <hip/hip_runtime.h>
#include <hip/hip_bf16.h>

typedef __attribute__((ext_vector_type(2))) float v2f;
typedef __attribute__((ext_vector_type(8))) float v8f;

static __device__ __forceinline__ v8f wmma_f32_16x16x4(v2f a, v2f b, v8f c) {
  // 8 args: (neg_a, A, neg_b, B, c_mod, C, reuse_a, reuse_b)
  return __builtin_amdgcn_wmma_f32_16x16x4_f32(false, a, false, b, (short)0, c,
                                               false, false);
}

// ---------------------------------------------------------------------------
// Zero a float region (grid-stride).
// ---------------------------------------------------------------------------
__global__ void zero_kernel(float* __restrict__ p, size_t n) {
  size_t i = (size_t)blockIdx.x * blockDim.x + threadIdx.x;
  size_t stride = (size_t)gridDim.x * blockDim.x;
  for (; i < n; i += stride) p[i] = 0.0f;
}

// ---------------------------------------------------------------------------
// Encoder: H_out[r, 0:32] = relu(X[r, 0:Kdim] @ W[Kdim,32] + bias[32])
// One wave per 16-row tile; fp32 WMMA 16x16x4, K stepped by 4, two N tiles.
// ---------------------------------------------------------------------------
__global__ __launch_bounds__(256) void encode_kernel(
    const float* __restrict__ X, const float* __restrict__ W,
    const float* __restrict__ bias, float* __restrict__ H_out,
    int n_rows, int Kdim) {
  const int lane = threadIdx.x & 31;
  const int wave = threadIdx.x >> 5;
  const int tile = blockIdx.x * (blockDim.x >> 5) + wave;
  const int row0 = tile * 16;
  if (row0 >= n_rows) return;  // wave-uniform exit, EXEC stays full below

  const int m  = lane & 15;         // M for A, N for B/C
  const int kb = (lane >> 4) * 2;   // K sub-offset inside the 4-wide step
  // clamp A-row for (non-occurring) ragged tails; per-lane select, no branch
  const int rA = (row0 + m < n_rows) ? (row0 + m) : (n_rows - 1);
  const float* xrow = X + (size_t)rA * Kdim;

  const int Mo = (lane >> 4) * 8;   // C/D: element i -> row Mo + i

  for (int n0 = 0; n0 < 32; n0 += 16) {
    v8f c;
    const float bv = bias[n0 + m];
#pragma unroll
    for (int i = 0; i < 8; ++i) c[i] = bv;

    for (int k0 = 0; k0 < Kdim; k0 += 4) {
      v2f a, b;
      a.x = xrow[k0 + kb + 0];
      a.y = xrow[k0 + kb + 1];
      const float* wr = W + (size_t)(k0 + kb) * 32 + n0 + m;
      b.x = wr[0];
      b.y = wr[32];
      c = wmma_f32_16x16x4(a, b, c);
    }
#pragma unroll
    for (int i = 0; i < 8; ++i) {
      const int r = row0 + Mo + i;
      if (r < n_rows) {
        float v = c[i];
        H_out[(size_t)r * 32 + n0 + m] = v > 0.0f ? v : 0.0f;
      }
    }
  }
}

// ---------------------------------------------------------------------------
// Scatter: one wave per edge. Coalesced 128B gather of the source row,
// lane-parallel f32 atomics into the destination accumulator (L2-resident).
// ---------------------------------------------------------------------------
__global__ __launch_bounds__(256) void scatter_kernel(
    const float* __restrict__ Hsrc, const int* __restrict__ src,
    const int* __restrict__ dst, float* __restrict__ sum,
    float* __restrict__ cnt, int n_edges) {
  const int lane = threadIdx.x & 31;
  const int e = (int)(((size_t)blockIdx.x * blockDim.x + threadIdx.x) >> 5);
  if (e >= n_edges) return;
  const int s = src[e];
  const int d = dst[e];
  const float v = Hsrc[(size_t)s * 32 + lane];
  atomicAdd(&sum[(size_t)d * 32 + lane], v);
  if (lane == 0) atomicAdd(&cnt[d], 1.0f);
}

// ---------------------------------------------------------------------------
// SAGE combine on bridge nodes (one wave per 16-row tile):
//   relu_out = relu( (sum_rb/cnt_rb)@Wl_rb + bl_rb
//                  + (sum_bb/cnt_bb)@Wl_bb + bl_bb
//                  +  h_b @ (Wr_rb + Wr_bb) )
// 3 fp32 WMMAs per K-step, 8 K-steps, 2 N-tiles = 48 WMMAs per tile.
// ---------------------------------------------------------------------------
__global__ __launch_bounds__(256) void sage_kernel(
    const float* __restrict__ h_b, const float* __restrict__ sum_rb,
    const float* __restrict__ cnt_rb, const float* __restrict__ sum_bb,
    const float* __restrict__ cnt_bb, const float* __restrict__ Wl_rb,
    const float* __restrict__ bl_rb, const float* __restrict__ Wr_rb,
    const float* __restrict__ Wl_bb, const float* __restrict__ bl_bb,
    const float* __restrict__ Wr_bb, float* __restrict__ relu_out, int nb) {
  const int lane = threadIdx.x & 31;
  const int wave = threadIdx.x >> 5;
  const int tile = blockIdx.x * (blockDim.x >> 5) + wave;
  const int row0 = tile * 16;
  if (row0 >= nb) return;

  const int m  = lane & 15;
  const int kb = (lane >> 4) * 2;
  const int rA = (row0 + m < nb) ? (row0 + m) : (nb - 1);

  const float inv_rb = 1.0f / fmaxf(cnt_rb[rA], 1.0f);
  const float inv_bb = 1.0f / fmaxf(cnt_bb[rA], 1.0f);

  const float* srb = sum_rb + (size_t)rA * 32;
  const float* sbb = sum_bb + (size_t)rA * 32;
  const float* hbr = h_b   + (size_t)rA * 32;

  const int Mo = (lane >> 4) * 8;

  for (int n0 = 0; n0 < 32; n0 += 16) {
    v8f c;
    const float bv = bl_rb[n0 + m] + bl_bb[n0 + m];
#pragma unroll
    for (int i = 0; i < 8; ++i) c[i] = bv;

    for (int k0 = 0; k0 < 32; k0 += 4) {
      const int k = k0 + kb;
      const size_t wofs = (size_t)k * 32 + n0 + m;

      v2f a, b;
      // mean_rb @ Wl_rb
      a.x = srb[k + 0] * inv_rb;
      a.y = srb[k + 1] * inv_rb;
      b.x = Wl_rb[wofs];
      b.y = Wl_rb[wofs + 32];
      c = wmma_f32_16x16x4(a, b, c);
      // mean_bb @ Wl_bb
      a.x = sbb[k + 0] * inv_bb;
      a.y = sbb[k + 1] * inv_bb;
      b.x = Wl_bb[wofs];
      b.y = Wl_bb[wofs + 32];
      c = wmma_f32_16x16x4(a, b, c);
      // h_b @ (Wr_rb + Wr_bb)   (lin_r has no bias)
      a.x = hbr[k + 0];
      a.y = hbr[k + 1];
      b.x = Wr_rb[wofs]      + Wr_bb[wofs];
      b.y = Wr_rb[wofs + 32] + Wr_bb[wofs + 32];
      c = wmma_f32_16x16x4(a, b, c);
    }
#pragma unroll
    for (int i = 0; i < 8; ++i) {
      const int r = row0 + Mo + i;
      if (r < nb) {
        float v = c[i];
        relu_out[(size_t)r * 32 + n0 + m] = v > 0.0f ? v : 0.0f;
      }
    }
  }
}

// ---------------------------------------------------------------------------
// Output head: out[r] = relu_out[r, :] . Wo + bo  (wave-per-node, shfl reduce)
// ---------------------------------------------------------------------------
__global__ __launch_bounds__(256) void head_kernel(
    const float* __restrict__ relu_out, const float* __restrict__ Wo,
    const float* __restrict__ bo, float* __restrict__ out, int nb) {
  const int lane = threadIdx.x & 31;
  const int node = (int)(((size_t)blockIdx.x * blockDim.x + threadIdx.x) >> 5);
  if (node >= nb) return;
  float v = relu_out[(size_t)node * 32 + lane] * Wo[lane];
#pragma unroll
  for (int off = 16; off > 0; off >>= 1) v += __shfl_xor(v, off, 32);
  if (lane == 0) out[node] = v + bo[0];
}

// ---------------------------------------------------------------------------
extern "C" void kernel_launch(void* const* d_in, const int* in_sizes, int n_in,
                              void* d_out, int out_size, void* d_ws,
                              size_t ws_size, hipStream_t stream) {
  const float* x_bridge = (const float*)d_in[0];
  const float* x_road   = (const float*)d_in[1];
  // d_in[2], d_in[3] (bridge->road edges) are dead code w.r.t. the output
  const int* src_rb = (const int*)d_in[4];
  const int* dst_rb = (const int*)d_in[5];
  const int* src_bb = (const int*)d_in[6];
  const int* dst_bb = (const int*)d_in[7];
  const float* W_enc_b = (const float*)d_in[8];
  const float* b_enc_b = (const float*)d_in[9];
  const float* W_enc_r = (const float*)d_in[10];
  const float* b_enc_r = (const float*)d_in[11];
  const float* Wl_rb = (const float*)d_in[15];
  const float* bl_rb = (const float*)d_in[16];
  const float* Wr_rb = (const float*)d_in[17];
  const float* Wl_bb = (const float*)d_in[18];
  const float* bl_bb = (const float*)d_in[19];
  const float* Wr_bb = (const float*)d_in[20];
  const float* Wo = (const float*)d_in[21];
  const float* bo = (const float*)d_in[22];
  float* out = (float*)d_out;

  const int NB = in_sizes[0] / 16;  // 100000
  const int NR = in_sizes[1] / 8;   // 200000
  const int Erb = in_sizes[4];      // 1000000
  const int Ebb = in_sizes[6];      // 500000

  // Workspace layout (floats). relu_out aliases h_r: h_r is dead after the
  // rb scatter, which completes (stream order) before sage_kernel writes it.
  // sum_rb, sum_bb, cnt_rb, cnt_bb are contiguous so one zero pass covers all.
  float* ws = (float*)d_ws;
  float* h_b      = ws;                       // NB*32 floats
  float* h_r      = h_b + (size_t)NB * 32;    // NR*32 floats (reused below)
  float* sum_rb   = h_r + (size_t)NR * 32;    // NB*32 floats (zeroed)
  float* sum_bb   = sum_rb + (size_t)NB * 32; // NB*32 floats (zeroed)
  float* cnt_rb   = sum_bb + (size_t)NB * 32; // NB floats (zeroed)
  float* cnt_bb   = cnt_rb + (size_t)NB;      // NB floats (zeroed)
  float* relu_out = h_r;

  const size_t zero_n = (size_t)NB * 64 + (size_t)NB * 2;
  zero_kernel<<<1024, 256, 0, stream>>>(sum_rb, zero_n);

  const int waves_per_block = 256 / 32;
  const int tiles_b = (NB + 15) / 16;
  const int tiles_r = (NR + 15) / 16;

  encode_kernel<<<(tiles_b + waves_per_block - 1) / waves_per_block, 256, 0,
                  stream>>>(x_bridge, W_enc_b, b_enc_b, h_b, NB, 16);
  encode_kernel<<<(tiles_r + waves_per_block - 1) / waves_per_block, 256, 0,
                  stream>>>(x_road, W_enc_r, b_enc_r, h_r, NR, 8);

  scatter_kernel<<<(Erb + waves_per_block - 1) / waves_per_block, 256, 0,
                   stream>>>(h_r, src_rb, dst_rb, sum_rb, cnt_rb, Erb);
  scatter_kernel<<<(Ebb + waves_per_block - 1) / waves_per_block, 256, 0,
                   stream>>>(h_b, src_bb, dst_bb, sum_bb, cnt_bb, Ebb);

  sage_kernel<<<(tiles_b + waves_per_block - 1) / waves_per_block, 256, 0,
                stream>>>(h_b, sum_rb, cnt_rb, sum_bb, cnt_bb, Wl_rb, bl_rb,
                          Wr_rb, Wl_bb, bl_bb, Wr_bb, relu_out, NB);

  head_kernel<<<(NB + waves_per_block - 1) / waves_per_block, 256, 0, stream>>>(
      relu_out, Wo, bo, out, NB);
}